// ADC_55465207660705
// MI455X (gfx1250) — compile-verified
//
#include <hip/hip_runtime.h>

// ADC LUT-interpolation + quantization, memory-bound elementwise kernel.
// HBM traffic: 512MB in + 512MB out -> ~43us floor at 23.3 TB/s.
// Strategy: async-stage LUT into LDS (CDNA5 global_load_async_to_lds_b128),
// uniform-grid index + fixup, LDS gathers, nontemporal b128 streaming.

typedef float v4f __attribute__((ext_vector_type(4)));

#define LUT_CAP 4096
#define NTHREADS 256

__global__ __launch_bounds__(NTHREADS) void ADC_55465207660705_kernel(
    const float* __restrict__ x,
    const float* __restrict__ lut_x,
    const float* __restrict__ lut_y,
    float* __restrict__ out,
    long long n, int lut_n)
{
    __shared__ float s_lx[LUT_CAP];   // xs table
    __shared__ float s_ly[LUT_CAP];   // ys table
    __shared__ float s_is[LUT_CAP];   // 1/(xs[i+1]-xs[i])
    __shared__ float s_red[NTHREADS / 32];

    const int t = threadIdx.x;

    // ---- Stage LUT into LDS ----
    if (lut_n == LUT_CAP) {
        // CDNA5 async global->LDS DMA path. Low 32 bits of a generic pointer
        // to __shared__ data are the raw LDS byte offset (aperture mapping).
        unsigned lx_base = (unsigned)(unsigned long long)(void*)&s_lx[0];
        unsigned ly_base = (unsigned)(unsigned long long)(void*)&s_ly[0];
        unsigned long long gx = (unsigned long long)lut_x;
        unsigned long long gy = (unsigned long long)lut_y;
        // 16KB per table, 256 lanes * 16B = 4KB per pass -> 4 passes each.
        #pragma unroll
        for (int k = 0; k < 4; ++k) {
            unsigned voff = (unsigned)(t * 16 + k * 4096);
            unsigned ldsx = lx_base + voff;
            unsigned ldsy = ly_base + voff;
            asm volatile("global_load_async_to_lds_b128 %0, %1, %2"
                         :: "v"(ldsx), "v"(voff), "s"(gx) : "memory");
            asm volatile("global_load_async_to_lds_b128 %0, %1, %2"
                         :: "v"(ldsy), "v"(voff), "s"(gy) : "memory");
        }
        asm volatile("s_wait_asynccnt 0" ::: "memory");
    } else {
        for (int i = t; i < lut_n; i += NTHREADS) {
            s_lx[i] = lut_x[i];
            s_ly[i] = lut_y[i];
        }
    }
    __syncthreads();

    // ---- Precompute inverse segment widths + block-wide max(lut_y) ----
    float lmax = -__builtin_inff();
    for (int i = t; i < lut_n - 1; i += NTHREADS)
        s_is[i] = 1.0f / (s_lx[i + 1] - s_lx[i]);
    for (int i = t; i < lut_n; i += NTHREADS)
        lmax = fmaxf(lmax, s_ly[i]);
    #pragma unroll
    for (int off = 16; off > 0; off >>= 1)            // wave32 reduction
        lmax = fmaxf(lmax, __shfl_xor(lmax, off, 32));
    if ((t & 31) == 0) s_red[t >> 5] = lmax;
    __syncthreads();
    float omax = s_red[0];
    #pragma unroll
    for (int i = 1; i < NTHREADS / 32; ++i) omax = fmaxf(omax, s_red[i]);

    const float xmin     = s_lx[0];
    const float xmax     = s_lx[lut_n - 1];
    const float vscale   = xmax * (1.0f / 8.0f);          // x/RELU_RANGE*adc_in_max
    const float inv_step = (float)(lut_n - 1) / (xmax - xmin);
    const float qscale   = 256.0f / omax;                 // full_scale / adc_out_max
    const float oscale   = 8.0f / 256.0f;                 // RELU_RANGE / full_scale
    const int   idx_hi   = lut_n - 2;

    auto proc = [&](float xi) -> float {
        float v = fminf(fmaxf(xi * vscale, xmin), xmax);
        int idx = (int)floorf((v - xmin) * inv_step);
        idx = idx < 0 ? 0 : (idx > idx_hi ? idx_hi : idx);
        // fixup against actual stored grid (ulp-safe at segment boundaries)
        if (idx > 0 && v < s_lx[idx])                --idx;
        else if (idx < idx_hi && v >= s_lx[idx + 1]) ++idx;
        float x0 = s_lx[idx];
        float y0 = s_ly[idx];
        float tt = (v - x0) * s_is[idx];
        float y  = fmaf(tt, s_ly[idx + 1] - y0, y0);
        return floorf(y * qscale) * oscale;
    };

    // ---- Streaming main loop: nontemporal b128 in/out ----
    const long long n4     = n >> 2;
    const long long stride = (long long)gridDim.x * NTHREADS;
    const long long base   = (long long)blockIdx.x * NTHREADS + t;

    const v4f* __restrict__ x4   = (const v4f*)x;
    v4f* __restrict__       out4 = (v4f*)out;

    for (long long i = base; i < n4; i += stride) {
        v4f xv = __builtin_nontemporal_load(x4 + i);
        v4f r;
        r.x = proc(xv.x);
        r.y = proc(xv.y);
        r.z = proc(xv.z);
        r.w = proc(xv.w);
        __builtin_nontemporal_store(r, out4 + i);
    }
    // scalar tail (n not multiple of 4)
    for (long long i = (n4 << 2) + base; i < n; i += stride) {
        float xi = __builtin_nontemporal_load(x + i);
        float r  = proc(xi);
        __builtin_nontemporal_store(r, out + i);
    }
}

extern "C" void kernel_launch(void* const* d_in, const int* in_sizes, int n_in,
                              void* d_out, int out_size, void* d_ws, size_t ws_size,
                              hipStream_t stream) {
    const float* x     = (const float*)d_in[0];
    const float* lut_x = (const float*)d_in[1];
    const float* lut_y = (const float*)d_in[2];
    float*       out   = (float*)d_out;

    long long n     = (long long)in_sizes[0];
    int       lut_n = (n_in > 1) ? in_sizes[1] : LUT_CAP;
    if (lut_n > LUT_CAP) lut_n = LUT_CAP;

    long long n4 = (n + 3) >> 2;
    long long want = (n4 + NTHREADS - 1) / NTHREADS;
    int blocks = (int)(want < 16384 ? (want < 1 ? 1 : want) : 16384);

    ADC_55465207660705_kernel<<<blocks, NTHREADS, 0, stream>>>(
        x, lut_x, lut_y, out, n, lut_n);
}